// SCHNETLayer_46102178955282
// MI455X (gfx1250) — compile-verified
//
#include <hip/hip_runtime.h>

#define H 128
#define HP 132                      // padded LDS row stride in dwords (bank-conflict-free)
#define TILE_ROWS 32                // rows per wave (two 16-row WMMA tiles share B fragments)
#define WAVES 4
#define BLOCK (WAVES * 32)                   // 128 threads
#define ROWS_PER_BLOCK (WAVES * TILE_ROWS)   // 128

typedef float v2f __attribute__((ext_vector_type(2)));
typedef float v8f __attribute__((ext_vector_type(8)));

// shifted softplus: softplus(x) - log(2), numerically stable
__device__ __forceinline__ float ssp_act(float x) {
    float t = __expf(-fabsf(x));
    return fmaxf(x, 0.0f) + __logf(1.0f + t) - 0.69314718055994530942f;
}

// Stage a 128x128 row-major weight into LDS with padded stride HP.
// 128 threads; each group of 32 lanes loads full rows (coalesced 512B/row).
__device__ __forceinline__ void stage_weight(const float* __restrict__ W,
                                             float* __restrict__ sW) {
    const int t = threadIdx.x;
    const int l = t & 31;
    const int g = t >> 5;
    for (int i = 0; i < H / WAVES; ++i) {
        const int r = g + i * WAVES;
        float4 v = reinterpret_cast<const float4*>(W + (size_t)r * H)[l];
        reinterpret_cast<float4*>(sW + r * HP)[l] = v;
    }
}

// One wave computes a 32x128 output tile: D = A(32x128) * W^T(128x128) + bias.
// Two 16-row A tiles (acc0 rows 0-15, acc1 rows 16-31) share every B fragment,
// halving LDS B-read traffic per WMMA.
// f32 WMMA 16x16x4: A frag lane(m=lane&15) holds K = k0 + 2*(lane>>4) + {0,1};
// B frag lane(n=lane&15) holds same K pair.  C/D: col = lane&15, row = v + 8*(lane>>4).
__device__ __forceinline__ void wave_gemm(const float* __restrict__ sXw,
                                          const float* __restrict__ sW,
                                          const float* __restrict__ sBias,
                                          int lane, v8f acc0[8], v8f acc1[8]) {
    const int lo = lane & 15;
    const int kh = (lane >> 4) << 1;
#pragma unroll
    for (int ct = 0; ct < 8; ++ct) {
        float bv = sBias[ct * 16 + lo];
        v8f a;
#pragma unroll
        for (int i = 0; i < 8; ++i) a[i] = bv;
        acc0[ct] = a;
        acc1[ct] = a;
    }
    const float* a0p = sXw + lo * HP + kh;
    const float* a1p = sXw + (16 + lo) * HP + kh;
    for (int k0 = 0; k0 < H; k0 += 4) {
        v2f av0 = *reinterpret_cast<const v2f*>(a0p + k0);
        v2f av1 = *reinterpret_cast<const v2f*>(a1p + k0);
#pragma unroll
        for (int ct = 0; ct < 8; ++ct) {
            v2f bv = *reinterpret_cast<const v2f*>(sW + (ct * 16 + lo) * HP + k0 + kh);
            acc0[ct] = __builtin_amdgcn_wmma_f32_16x16x4_f32(
                false, av0, false, bv, (short)0, acc0[ct], false, false);
            acc1[ct] = __builtin_amdgcn_wmma_f32_16x16x4_f32(
                false, av1, false, bv, (short)0, acc1[ct], false, false);
        }
    }
}

// ---------------- kernel 1: hn = h @ W_node^T + b_node ----------------
__global__ __launch_bounds__(BLOCK) void node_gemm_kernel(
    const float* __restrict__ h, const float* __restrict__ Wn,
    const float* __restrict__ bn, float* __restrict__ hn, int N) {
    __shared__ float sW[H * HP];
    __shared__ float sX[WAVES][TILE_ROWS * HP];
    __shared__ float sB[H];
    const int t = threadIdx.x, lane = t & 31, wave = t >> 5;
    sB[t] = bn[t];                      // BLOCK == H
    stage_weight(Wn, sW);
    const int base = blockIdx.x * ROWS_PER_BLOCK + wave * TILE_ROWS;
#pragma unroll
    for (int r = 0; r < TILE_ROWS; ++r) {
        const int row = base + r;
        float4 v = make_float4(0.f, 0.f, 0.f, 0.f);
        if (row < N) v = reinterpret_cast<const float4*>(h + (size_t)row * H)[lane];
        reinterpret_cast<float4*>(&sX[wave][r * HP])[lane] = v;
    }
    __syncthreads();
    v8f acc0[8], acc1[8];
    wave_gemm(sX[wave], sW, sB, lane, acc0, acc1);
    const int hi = lane >> 4, lo = lane & 15;
#pragma unroll
    for (int ct = 0; ct < 8; ++ct) {
#pragma unroll
        for (int v = 0; v < 8; ++v) {
            const int r0 = base + v + 8 * hi;
            const int r1 = r0 + 16;
            if (r0 < N) hn[(size_t)r0 * H + ct * 16 + lo] = acc0[ct][v];
            if (r1 < N) hn[(size_t)r1 * H + ct * 16 + lo] = acc1[ct][v];
        }
    }
}

// ---- kernel 2 (fused): ef = ssp(ssp(e@W1^T+b1)@W2^T+b2); atomically agg[dst] += hn[src]*ef ----
__global__ __launch_bounds__(BLOCK) void edge_mlp_kernel(
    const float* __restrict__ e, const float* __restrict__ W1, const float* __restrict__ b1,
    const float* __restrict__ W2, const float* __restrict__ b2,
    const float* __restrict__ hn, const int* __restrict__ src, const int* __restrict__ dst,
    float* __restrict__ agg) {
    __shared__ float sW[H * HP];
    __shared__ float sX[WAVES][TILE_ROWS * HP];
    __shared__ float sB1[H];
    __shared__ float sB2[H];
    const int t = threadIdx.x, lane = t & 31, wave = t >> 5;
    sB1[t] = b1[t];
    sB2[t] = b2[t];
    stage_weight(W1, sW);
    const int ebase = blockIdx.x * ROWS_PER_BLOCK + wave * TILE_ROWS;  // E % 128 == 0
#pragma unroll
    for (int r = 0; r < TILE_ROWS; ++r) {
        float4 v = reinterpret_cast<const float4*>(e + (size_t)(ebase + r) * H)[lane];
        reinterpret_cast<float4*>(&sX[wave][r * HP])[lane] = v;
    }
    __syncthreads();
    v8f acc0[8], acc1[8];
    wave_gemm(sX[wave], sW, sB1, lane, acc0, acc1);
    const int hi = lane >> 4, lo = lane & 15;
    // ssp, write activated tile back over the wave-private A tile
#pragma unroll
    for (int ct = 0; ct < 8; ++ct) {
#pragma unroll
        for (int v = 0; v < 8; ++v) {
            sX[wave][(v + 8 * hi) * HP + ct * 16 + lo]      = ssp_act(acc0[ct][v]);
            sX[wave][(16 + v + 8 * hi) * HP + ct * 16 + lo] = ssp_act(acc1[ct][v]);
        }
    }
    __syncthreads();              // everyone done reading W1
    stage_weight(W2, sW);         // reuse LDS buffer -> total LDS ~136 KB, 2 WGs/WGP
    __syncthreads();
    wave_gemm(sX[wave], sW, sB2, lane, acc0, acc1);
    int sidx[16], didx[16];
#pragma unroll
    for (int v = 0; v < 8; ++v) {
        const int er = ebase + v + 8 * hi;
        sidx[v]     = src[er];
        didx[v]     = dst[er];
        sidx[v + 8] = src[er + 16];
        didx[v + 8] = dst[er + 16];
    }
#pragma unroll
    for (int ct = 0; ct < 8; ++ct) {
#pragma unroll
        for (int v = 0; v < 8; ++v) {
            const int c = ct * 16 + lo;
            const float val0 = ssp_act(acc0[ct][v]);
            const float val1 = ssp_act(acc1[ct][v]);
            const float hv0 = hn[(size_t)sidx[v] * H + c];
            const float hv1 = hn[(size_t)sidx[v + 8] * H + c];
            atomicAdd(agg + (size_t)didx[v] * H + c, hv0 * val0);
            atomicAdd(agg + (size_t)didx[v + 8] * H + c, hv1 * val1);
        }
    }
}

// ---- kernel 3 (fused): out = ssp(agg@Wfc1^T+b)@Wfc2^T+b ----
__global__ __launch_bounds__(BLOCK) void ffn_kernel(
    const float* __restrict__ agg, const float* __restrict__ W1, const float* __restrict__ b1,
    const float* __restrict__ W2, const float* __restrict__ b2,
    float* __restrict__ out, int N) {
    __shared__ float sW[H * HP];
    __shared__ float sX[WAVES][TILE_ROWS * HP];
    __shared__ float sB1[H];
    __shared__ float sB2[H];
    const int t = threadIdx.x, lane = t & 31, wave = t >> 5;
    sB1[t] = b1[t];
    sB2[t] = b2[t];
    stage_weight(W1, sW);
    const int base = blockIdx.x * ROWS_PER_BLOCK + wave * TILE_ROWS;
#pragma unroll
    for (int r = 0; r < TILE_ROWS; ++r) {
        const int row = base + r;
        float4 v = make_float4(0.f, 0.f, 0.f, 0.f);
        if (row < N) v = reinterpret_cast<const float4*>(agg + (size_t)row * H)[lane];
        reinterpret_cast<float4*>(&sX[wave][r * HP])[lane] = v;
    }
    __syncthreads();
    v8f acc0[8], acc1[8];
    wave_gemm(sX[wave], sW, sB1, lane, acc0, acc1);
    const int hi = lane >> 4, lo = lane & 15;
#pragma unroll
    for (int ct = 0; ct < 8; ++ct) {
#pragma unroll
        for (int v = 0; v < 8; ++v) {
            sX[wave][(v + 8 * hi) * HP + ct * 16 + lo]      = ssp_act(acc0[ct][v]);
            sX[wave][(16 + v + 8 * hi) * HP + ct * 16 + lo] = ssp_act(acc1[ct][v]);
        }
    }
    __syncthreads();
    stage_weight(W2, sW);
    __syncthreads();
    wave_gemm(sX[wave], sW, sB2, lane, acc0, acc1);
#pragma unroll
    for (int ct = 0; ct < 8; ++ct) {
#pragma unroll
        for (int v = 0; v < 8; ++v) {
            const int r0 = base + v + 8 * hi;
            const int r1 = r0 + 16;
            if (r0 < N) out[(size_t)r0 * H + ct * 16 + lo] = acc0[ct][v];
            if (r1 < N) out[(size_t)r1 * H + ct * 16 + lo] = acc1[ct][v];
        }
    }
}

extern "C" void kernel_launch(void* const* d_in, const int* in_sizes, int n_in,
                              void* d_out, int out_size, void* d_ws, size_t ws_size,
                              hipStream_t stream) {
    const float* h   = (const float*)d_in[0];
    const float* e   = (const float*)d_in[1];
    const float* Wn  = (const float*)d_in[2];
    const float* bn  = (const float*)d_in[3];
    const float* W1  = (const float*)d_in[4];
    const float* b1  = (const float*)d_in[5];
    const float* W2  = (const float*)d_in[6];
    const float* b2  = (const float*)d_in[7];
    const float* Wf1 = (const float*)d_in[8];
    const float* bf1 = (const float*)d_in[9];
    const float* Wf2 = (const float*)d_in[10];
    const float* bf2 = (const float*)d_in[11];
    const int*   src = (const int*)d_in[12];
    const int*   dst = (const int*)d_in[13];

    const int N = in_sizes[0] / H;
    const int E = in_sizes[12];

    float* hn  = (float*)d_ws;                 // [N,128]
    float* agg = hn + (size_t)N * H;           // [N,128]

    hipMemsetAsync(agg, 0, (size_t)N * H * sizeof(float), stream);

    const int nb = (N + ROWS_PER_BLOCK - 1) / ROWS_PER_BLOCK;
    node_gemm_kernel<<<nb, BLOCK, 0, stream>>>(h, Wn, bn, hn, N);
    edge_mlp_kernel<<<E / ROWS_PER_BLOCK, BLOCK, 0, stream>>>(e, W1, b1, W2, b2,
                                                              hn, src, dst, agg);
    ffn_kernel<<<nb, BLOCK, 0, stream>>>(agg, Wf1, bf1, Wf2, bf2, (float*)d_out, N);
}